// PiCoPixelContrastLoss_21715354649036
// MI455X (gfx1250) — compile-verified
//
#include <hip/hip_runtime.h>
#include <math.h>

typedef __attribute__((ext_vector_type(2))) float v2f;
typedef __attribute__((ext_vector_type(8))) float v8f;

#define NUM_CLASSES 19
#define KPC        100
#define NA         (NUM_CLASSES * KPC)   // 1900 anchors
#define NMEM       20000
#define NC         (NA + NMEM)           // 21900 real contrast rows
#define NCP        21952                 // padded to multiple of 64 (zero rows)
#define DIM        256
#define HW         16384                 // 128*128
#define NL         131072                // 8*128*128 labels
#define TEMP_INV   10.0f                 // 1/0.1
#define LOSS_SCALE (0.1f / 0.07f)

#define WAVES      8
#define NSPLIT     8
#define NSTREAM    (WAVES * NSPLIT)      // 64 column streams
#define NTILEP     (NCP / 16)            // 1372 column tiles (padded)
#define NQUADS     (NTILEP / 4)          // 343 tile-quads (64 cols each)
#define NBLK       ((NA + 15) / 16)      // 119 anchor blocks
#define LDSSTRIDE  260                   // padded row stride (floats), conflict-free B64 reads

// ---------------------------------------------------------------------------
// Kernel 1: deterministic "first K indices per class" sampler.
// One wave per class; ordered chunk scan with ballot/popc prefix ranks.
// ---------------------------------------------------------------------------
__global__ __launch_bounds__(32) void sample_kernel(const int* __restrict__ labels,
                                                    int* __restrict__ idx) {
    int c    = blockIdx.x;          // class
    int lane = threadIdx.x;         // 0..31 (wave32)

    for (int i = lane; i < KPC; i += 32) idx[c * KPC + i] = 0;  // match fill_value=0

    int count = 0;
    for (int base = 0; base < NL && count < KPC; base += 32) {
        int i = base + lane;
        bool m = (i < NL) && (labels[i] == c);
        unsigned long long bal = __ballot(m);
        unsigned mask = (unsigned)bal;                      // wave32: low 32 bits
        int rank = __popc(mask & ((1u << lane) - 1u));
        if (m && (count + rank) < KPC) idx[c * KPC + count + rank] = i;
        count += __popc(mask);
    }
}

// ---------------------------------------------------------------------------
// Kernel 2: build normalized contrast matrix Cn[NCP][DIM] and labels cl[NCP].
// Rows 0..NA-1: gathered anchors (strided NCHW gather), NA..NC-1: memory,
// NC..NCP-1: zero padding (label -3) so the GEMM needs no in-loop masking.
// ---------------------------------------------------------------------------
__global__ __launch_bounds__(256) void build_contrast(const float* __restrict__ pf,
                                                      const int*   __restrict__ idx,
                                                      const float* __restrict__ mf,
                                                      const int*   __restrict__ ml,
                                                      float* __restrict__ Cn,
                                                      int*   __restrict__ cl) {
    int row = blockIdx.x;           // 0..NCP-1
    int d   = threadIdx.x;          // 0..255

    float v = 0.0f;
    if (row < NA) {
        int p  = idx[row];          // flat pixel index in [0, NL)
        int b  = p >> 14;           // / (128*128)
        int hw = p & (HW - 1);
        v = pf[(((size_t)b * DIM + d) << 14) + hw];   // pixel_feats[b][d][h][w]
    } else if (row < NC) {
        v = mf[(size_t)(row - NA) * DIM + d];
    }

    __shared__ float red[WAVES];
    float sq = v * v;
    #pragma unroll
    for (int o = 16; o > 0; o >>= 1) sq += __shfl_xor(sq, o, 32);
    if ((threadIdx.x & 31) == 0) red[threadIdx.x >> 5] = sq;
    __syncthreads();
    float tot = 0.0f;
    #pragma unroll
    for (int w = 0; w < WAVES; w++) tot += red[w];

    float rn = 1.0f / fmaxf(sqrtf(tot), 1e-12f);      // matches F.normalize eps
    Cn[(size_t)row * DIM + d] = v * rn;               // pad rows: 0 * 1e12 = 0
    if (d == 0) cl[row] = (row < NA) ? (row / KPC) : ((row < NC) ? ml[row - NA] : -3);
}

// ---------------------------------------------------------------------------
// Kernel 3: fused GEMM (V_WMMA_F32_16X16X4_F32) + masked-softmax statistics.
// grid = (NBLK anchor blocks, NSPLIT column splits), 256 threads = 8 waves.
// A tile staged in LDS; each wave streams QUADS of 16-column tiles with 4
// independent accumulator chains and software double-buffered B fragments:
// the 4 B loads for step k+1 are in flight across the 4 WMMAs of step k.
// ---------------------------------------------------------------------------
__global__ __launch_bounds__(256) void contrast_main(const float* __restrict__ Cn,
                                                     const int*   __restrict__ cl,
                                                     float* __restrict__ partS,
                                                     float* __restrict__ partP,
                                                     float* __restrict__ partC) {
    __shared__ float ldsA[16 * LDSSTRIDE];

    int n0 = blockIdx.x * 16;

    // Stage A tile (rows n0..n0+15 of Cn, zero-padded past NA) into LDS.
    for (int i = threadIdx.x; i < 16 * DIM; i += 256) {
        int r = i >> 8, c = i & (DIM - 1);
        float v = (n0 + r < NA) ? Cn[(size_t)(n0 + r) * DIM + c] : 0.0f;
        ldsA[r * LDSSTRIDE + c] = v;
    }
    __syncthreads();

    int wave = threadIdx.x >> 5;
    int lane = threadIdx.x & 31;
    int hi   = lane >> 4;           // lane half selects K-pair / M-half
    int nm   = lane & 15;           // A: M row; B/D: N column
    int s    = blockIdx.y * WAVES + wave;   // stream id 0..63

    float Sa[8], Pa[8], Ca[8];
    int rowlab[8];
    #pragma unroll
    for (int v = 0; v < 8; v++) {
        Sa[v] = 0.0f; Pa[v] = 0.0f; Ca[v] = 0.0f;
        int row = n0 + v + hi * 8;                     // D layout: M = v + 8*hi
        rowlab[v] = (row < NA) ? (row / KPC) : -2;
    }

    const int aoff = nm * LDSSTRIDE + 2 * hi;

    for (int q = s; q < NQUADS; q += NSTREAM) {
        int base = q * 64 + nm;                        // tiles 4q..4q+3
        const float* brow0 = Cn + (size_t)(base     ) * DIM + 2 * hi;
        const float* brow1 = Cn + (size_t)(base + 16) * DIM + 2 * hi;
        const float* brow2 = Cn + (size_t)(base + 32) * DIM + 2 * hi;
        const float* brow3 = Cn + (size_t)(base + 48) * DIM + 2 * hi;

        v8f ac0 = {0.0f, 0.0f, 0.0f, 0.0f, 0.0f, 0.0f, 0.0f, 0.0f};
        v8f ac1 = ac0, ac2 = ac0, ac3 = ac0;

        // Software pipeline: prime current B fragments, then each step issues
        // next-step loads BEFORE the WMMAs that consume the current ones.
        v2f cb0 = *(const v2f*)(brow0);
        v2f cb1 = *(const v2f*)(brow1);
        v2f cb2 = *(const v2f*)(brow2);
        v2f cb3 = *(const v2f*)(brow3);
        #pragma unroll 8
        for (int kk = 0; kk < DIM; kk += 4) {
            // Prefetch step kk+4 (final iteration harmlessly overreads 8B
            // into the adjacent workspace region; value is discarded).
            v2f nb0 = *(const v2f*)(brow0 + kk + 4);
            v2f nb1 = *(const v2f*)(brow1 + kk + 4);
            v2f nb2 = *(const v2f*)(brow2 + kk + 4);
            v2f nb3 = *(const v2f*)(brow3 + kk + 4);
            v2f a   = *(const v2f*)&ldsA[aoff + kk];   // A[M=nm][kk+2hi..]
            ac0 = __builtin_amdgcn_wmma_f32_16x16x4_f32(
                false, a, false, cb0, (short)0, ac0, false, false);
            ac1 = __builtin_amdgcn_wmma_f32_16x16x4_f32(
                false, a, false, cb1, (short)0, ac1, false, false);
            ac2 = __builtin_amdgcn_wmma_f32_16x16x4_f32(
                false, a, false, cb2, (short)0, ac2, false, false);
            ac3 = __builtin_amdgcn_wmma_f32_16x16x4_f32(
                false, a, false, cb3, (short)0, ac3, false, false);
            cb0 = nb0; cb1 = nb1; cb2 = nb2; cb3 = nb3;
        }

        auto epilogue = [&](const v8f& acc, int col) {
            int clab = (col < NC) ? cl[col] : -3;
            #pragma unroll
            for (int v = 0; v < 8; v++) {
                int row = n0 + v + hi * 8;
                float l = acc[v] * TEMP_INV;
                if (col < NC && col != row) {
                    Sa[v] += __expf(l);
                    if (clab == rowlab[v]) { Pa[v] += l; Ca[v] += 1.0f; }
                }
            }
        };
        epilogue(ac0, base);
        epilogue(ac1, base + 16);
        epilogue(ac2, base + 32);
        epilogue(ac3, base + 48);
    }

    // Reduce across the 16-lane half that shares each row (distinct columns).
    #pragma unroll
    for (int v = 0; v < 8; v++) {
        float sS = Sa[v], sP = Pa[v], sC = Ca[v];
        #pragma unroll
        for (int o = 1; o < 16; o <<= 1) {
            sS += __shfl_xor(sS, o, 32);
            sP += __shfl_xor(sP, o, 32);
            sC += __shfl_xor(sC, o, 32);
        }
        if (nm == 0) {                                 // lanes 0 and 16 write
            int row = n0 + v + hi * 8;
            if (row < NA) {
                partS[(size_t)s * NA + row] = sS;
                partP[(size_t)s * NA + row] = sP;
                partC[(size_t)s * NA + row] = sC;
            }
        }
    }
}

// ---------------------------------------------------------------------------
// Kernel 4: deterministic finalize. Sum the 64 per-stream partials in fixed
// order, form mean_log_prob_pos per row, reduce to the scalar loss.
// ---------------------------------------------------------------------------
__global__ __launch_bounds__(256) void finalize_kernel(const float* __restrict__ partS,
                                                       const float* __restrict__ partP,
                                                       const float* __restrict__ partC,
                                                       float* __restrict__ out) {
    __shared__ float shS[WAVES], shV[WAVES];
    float lsum = 0.0f, lval = 0.0f;

    for (int n = threadIdx.x; n < NA; n += 256) {
        float S = 0.0f, P = 0.0f, C = 0.0f;
        for (int si = 0; si < NSTREAM; si++) {
            S += partS[(size_t)si * NA + n];
            P += partP[(size_t)si * NA + n];
            C += partC[(size_t)si * NA + n];
        }
        if (C > 0.0f) {
            // sum_pos(logits) - cnt*log(S + 1e-12), divided by cnt
            lsum += (P - C * logf(S + 1e-12f)) / C;
            lval += 1.0f;
        }
    }
    #pragma unroll
    for (int o = 16; o > 0; o >>= 1) {
        lsum += __shfl_xor(lsum, o, 32);
        lval += __shfl_xor(lval, o, 32);
    }
    if ((threadIdx.x & 31) == 0) {
        shS[threadIdx.x >> 5] = lsum;
        shV[threadIdx.x >> 5] = lval;
    }
    __syncthreads();
    if (threadIdx.x == 0) {
        float S = 0.0f, V = 0.0f;
        #pragma unroll
        for (int i = 0; i < WAVES; i++) { S += shS[i]; V += shV[i]; }
        out[0] = -LOSS_SCALE * (S / fmaxf(V, 1.0f));
    }
}

// ---------------------------------------------------------------------------
extern "C" void kernel_launch(void* const* d_in, const int* in_sizes, int n_in,
                              void* d_out, int out_size, void* d_ws, size_t ws_size,
                              hipStream_t stream) {
    const float* pf     = (const float*)d_in[0];  // pixel_feats [8,256,128,128]
    const int*   labels = (const int*)  d_in[1];  // labels      [8,128,128]
    const float* mf     = (const float*)d_in[2];  // memory_feats [20000,256]
    const int*   ml     = (const int*)  d_in[3];  // memory_labels [20000]

    char* ws = (char*)d_ws;
    const size_t CN_BYTES  = (size_t)NCP * DIM * sizeof(float);      // ~22.5 MB
    const size_t IDX_BYTES = 8192;                                   // 1900 ints, padded
    const size_t CL_BYTES  = 90112;                                  // 21952 ints, padded

    float* Cn    = (float*)(ws);
    int*   idx   = (int*)  (ws + CN_BYTES);
    int*   cl    = (int*)  (ws + CN_BYTES + IDX_BYTES);
    float* partS = (float*)(ws + CN_BYTES + IDX_BYTES + CL_BYTES);
    float* partP = partS + (size_t)NSTREAM * NA;
    float* partC = partP + (size_t)NSTREAM * NA;
    (void)in_sizes; (void)n_in; (void)out_size; (void)ws_size;

    sample_kernel  <<<dim3(NUM_CLASSES), dim3(32),  0, stream>>>(labels, idx);
    build_contrast <<<dim3(NCP),         dim3(256), 0, stream>>>(pf, idx, mf, ml, Cn, cl);
    contrast_main  <<<dim3(NBLK, NSPLIT), dim3(256), 0, stream>>>(Cn, cl, partS, partP, partC);
    finalize_kernel<<<dim3(1),           dim3(256), 0, stream>>>(partS, partP, partC, (float*)d_out);
}